// UpSampleConst_36653250904491
// MI455X (gfx1250) — compile-verified
//
#include <hip/hip_runtime.h>

// out[c, j] = in[c, idx[j]]   (B=1 folded into C)
// in : [C, N_IN]  fp32   (128 MB  -> fits in 192 MB L2: keep resident, default RT loads)
// idx: [N_OUT]    i32    (16 MB   -> sequential, read once, prefetched)
// out: [C, N_OUT] fp32   (512 MB  -> streaming, NT b128 stores so the write stream
//                          never evicts the L2-resident input)

typedef float v4f __attribute__((ext_vector_type(4)));
typedef int   v4i __attribute__((ext_vector_type(4)));

template <int NCH>
__global__ __launch_bounds__(256) void UpSampleConst_gather_kernel(
    const float* __restrict__ in, const int* __restrict__ idx,
    float* __restrict__ out, int n_in, int n_out, int n_ch_rt)
{
    const int n_ch = (NCH > 0) ? NCH : n_ch_rt;

    const long long nvec   = (long long)(n_out >> 2);         // float4 groups of j
    const long long tid    = (long long)blockIdx.x * blockDim.x + threadIdx.x;
    const long long stride = (long long)gridDim.x * blockDim.x;

    for (long long v = tid; v < nvec; v += stride) {
        // Prefetch the next grid-stride chunk of the sequential index stream
        // (global_prefetch_b8; speculative, silently dropped if untranslatable).
        if (v + stride < nvec)
            __builtin_prefetch(idx + ((v + stride) << 2), 0, 1);

        const long long j = v << 2;
        const v4i p = *(const v4i*)(idx + j);                  // one coalesced b128 load

        const float* __restrict__ src = in;
        float*       __restrict__ dst = out + j;
#pragma unroll 8
        for (int c = 0; c < n_ch; ++c) {
            v4f r;
            r.x = src[p.x];                                    // random 4B gathers,
            r.y = src[p.y];                                    // L2-resident (RT)
            r.z = src[p.z];
            r.w = src[p.w];
            __builtin_nontemporal_store(r, (v4f*)dst);         // b128 th:NT stream-out
            src += n_in;
            dst += n_out;
        }
    }

    // Tail when n_out % 4 != 0 (not hit for N_OUT = 4M, kept for correctness).
    const long long tail = nvec << 2;
    for (long long j = tail + tid; j < n_out; j += stride) {
        const int p = idx[j];
        for (int c = 0; c < n_ch; ++c)
            __builtin_nontemporal_store(in[(long long)c * n_in + p],
                                        out + (long long)c * n_out + j);
    }
}

extern "C" void kernel_launch(void* const* d_in, const int* in_sizes, int n_in_arrs,
                              void* d_out, int out_size, void* d_ws, size_t ws_size,
                              hipStream_t stream) {
    const float* in  = (const float*)d_in[0];   // input_features [1, C, N_IN] fp32
    const int*   idx = (const int*)d_in[1];     // parent_idx [N_OUT] i32
    // d_in[2] (level_deltas) does not affect the computed output.
    float*       out = (float*)d_out;           // [1, C, N_OUT] fp32

    const int n_out = in_sizes[1];                                  // N_OUT
    const int n_ch  = (n_out > 0) ? (out_size / n_out) : 0;         // B*C (= 32)
    const int n_in  = (n_ch  > 0) ? (in_sizes[0] / n_ch) : 0;       // N_IN

    // ~2 grid-stride iterations per thread so the index prefetch overlaps real
    // work; ~500K threads is still ~15.6K wave32s — far beyond WGP residency.
    long long nvec = (long long)(n_out >> 2);
    long long want_threads = (nvec + 1) / 2;
    if (want_threads < 1) want_threads = 1;
    int blocks = (int)((want_threads + 255) / 256);
    if (blocks < 1) blocks = 1;

    if (n_ch == 32) {
        UpSampleConst_gather_kernel<32><<<blocks, 256, 0, stream>>>(
            in, idx, out, n_in, n_out, n_ch);
    } else {
        UpSampleConst_gather_kernel<0><<<blocks, 256, 0, stream>>>(
            in, idx, out, n_in, n_out, n_ch);
    }
}